// SddnSelect_56513179680800
// MI455X (gfx1250) — compile-verified
//
#include <hip/hip_runtime.h>
#include <hip/hip_bf16.h>

// Problem constants (fixed shapes from the reference)
#define BDIM 16
#define KCAND 16
#define CDIM 64
#define HW   4096                 // 64*64
#define CHW  262144               // 64*64*64 elems per (b,k)
#define CHW4 65536                // CHW / 4 (float4 units)
#define SPLIT 16
#define CHUNK4 4096               // CHW4 / SPLIT (float4 per chunk)
#define SEL_ELEMS 4194304         // B*C*H*W
#define PENALTY 0.0009765625f     // log2(16)/4096
#define INV_CHW (1.0f/262144.0f)

typedef __attribute__((ext_vector_type(2))) float v2f;
typedef __attribute__((ext_vector_type(4))) float v4f;
typedef __attribute__((ext_vector_type(8))) float v8f;

// Reduce one lane-private float across the wave32 using V_WMMA_F32_16X16X4_F32
// with a ones B-matrix.  Stage 1: D1[m,n] = acc[m] + acc[m+16].  Stage 2: four
// accumulating WMMAs over row pairs of D1 give D2[m,n] = sum of all 32 lanes,
// replicated in every element.  EXEC must be all ones (uniform flow here).
__device__ __forceinline__ float wave_sum_wmma(float acc) {
    v2f a;    a.x = acc;  a.y = 0.0f;
    v2f ones; ones.x = 1.0f; ones.y = 1.0f;
    v8f d = {};
    d = __builtin_amdgcn_wmma_f32_16x16x4_f32(false, a, false, ones, (short)0, d, false, false);
    v8f e = {};
    v2f p;
    p.x = d[0]; p.y = d[1];
    e = __builtin_amdgcn_wmma_f32_16x16x4_f32(false, p, false, ones, (short)0, e, false, false);
    p.x = d[2]; p.y = d[3];
    e = __builtin_amdgcn_wmma_f32_16x16x4_f32(false, p, false, ones, (short)0, e, false, false);
    p.x = d[4]; p.y = d[5];
    e = __builtin_amdgcn_wmma_f32_16x16x4_f32(false, p, false, ones, (short)0, e, false, false);
    p.x = d[6]; p.y = d[7];
    e = __builtin_amdgcn_wmma_f32_16x16x4_f32(false, p, false, ones, (short)0, e, false, false);
    return e[0];
}

// Pass 1: partial[bk*SPLIT + s] = sum over chunk s of (x[bk] - target[b])^2
__global__ __launch_bounds__(256)
void sddn_loss_partial(const float* __restrict__ x,
                       const float* __restrict__ target,
                       float* __restrict__ partial) {
    const int bk = blockIdx.x >> 4;       // 0..255  (b*K + k)
    const int s  = blockIdx.x & 15;       // chunk id
    const int b  = bk >> 4;
    const int tid = threadIdx.x;

    const v4f* __restrict__ x4 = (const v4f*)x;
    const v4f* __restrict__ t4 = (const v4f*)target;
    const long xbase = (long)bk * CHW4 + (long)s * CHUNK4;
    const long tbase = (long)b  * CHW4 + (long)s * CHUNK4;

    float acc = 0.0f;
#pragma unroll 4
    for (int i = tid; i < CHUNK4; i += 256) {
        // x is streamed once: non-temporal so it doesn't evict target from L2.
        v4f xv = __builtin_nontemporal_load(x4 + xbase + i);
        v4f tv = t4[tbase + i];           // target: regular temporal (16x reuse)
        float d0 = xv.x - tv.x;
        float d1 = xv.y - tv.y;
        float d2 = xv.z - tv.z;
        float d3 = xv.w - tv.w;
        acc = fmaf(d0, d0, acc);
        acc = fmaf(d1, d1, acc);
        acc = fmaf(d2, d2, acc);
        acc = fmaf(d3, d3, acc);
    }

    // wave32 reduction via WMMA (uniform control flow -> EXEC all ones)
    float wsum = wave_sum_wmma(acc);

    __shared__ float lds[8];
    const int lane = tid & 31;
    const int wv   = tid >> 5;
    if (lane == 0) lds[wv] = wsum;
    __syncthreads();
    if (tid == 0) {
        float tot = 0.0f;
#pragma unroll
        for (int w = 0; w < 8; ++w) tot += lds[w];
        partial[bk * SPLIT + s] = tot;
    }
}

// Pass 2: finish losses, weighted argmin per b, emit min_loss + idx
__global__ __launch_bounds__(256)
void sddn_argmin(const float* __restrict__ partial,
                 const float* __restrict__ pf,
                 int* __restrict__ idxbuf,
                 float* __restrict__ out) {
    __shared__ float loss_s[BDIM * KCAND];
    const int t = threadIdx.x;            // t = b*K + k
    float ssum = 0.0f;
#pragma unroll
    for (int i = 0; i < SPLIT; ++i) ssum += partial[t * SPLIT + i];  // fixed order
    loss_s[t] = ssum * INV_CHW + PENALTY;
    __syncthreads();
    if (t < BDIM) {                       // t = b
        float best = __builtin_inff();
        float bloss = 0.0f;
        int   bi = 0;
#pragma unroll
        for (int k = 0; k < KCAND; ++k) {
            float L = loss_s[t * KCAND + k];
            float w = L * pf[k];
            if (w < best) { best = w; bi = k; bloss = L; }  // strict < == first-min
        }
        idxbuf[t] = bi;
        out[SEL_ELEMS + t] = bloss;       // min_loss tail
    }
}

// Pass 3: gather selected candidate (float4 per thread, NT in and out)
__global__ __launch_bounds__(256)
void sddn_gather(const float* __restrict__ x,
                 const int* __restrict__ idxbuf,
                 float* __restrict__ out) {
    const int g = blockIdx.x * 256 + threadIdx.x;   // 0 .. 1048575 (float4 idx)
    const int b = g >> 16;                          // 65536 float4 per b
    const int i = g & 65535;
    const int k = idxbuf[b];                        // uniform within a block
    const v4f* __restrict__ x4 = (const v4f*)x;
    v4f v = __builtin_nontemporal_load(x4 + ((long)(b * KCAND + k) * CHW4 + i));
    __builtin_nontemporal_store(v, (v4f*)out + g);
}

extern "C" void kernel_launch(void* const* d_in, const int* in_sizes, int n_in,
                              void* d_out, int out_size, void* d_ws, size_t ws_size,
                              hipStream_t stream) {
    const float* x      = (const float*)d_in[0];
    const float* target = (const float*)d_in[1];
    const float* pf     = (const float*)d_in[2];
    float* out = (float*)d_out;

    float* partial = (float*)d_ws;                               // 4096 floats
    int*   idxbuf  = (int*)((char*)d_ws + 4096 * sizeof(float)); // 16 ints

    // Pass 1: 256 (b,k) pairs x 16 chunks = 4096 blocks
    sddn_loss_partial<<<BDIM * KCAND * SPLIT, 256, 0, stream>>>(x, target, partial);
    // Pass 2: one block finalizes losses and argmins
    sddn_argmin<<<1, 256, 0, stream>>>(partial, pf, idxbuf, out);
    // Pass 3: 4,194,304 floats / 4 / 256 = 4096 blocks
    sddn_gather<<<4096, 256, 0, stream>>>(x, idxbuf, out);
}